// LogSparsemaxBisect_62551903699249
// MI455X (gfx1250) — compile-verified
//
#include <hip/hip_runtime.h>
#include <math.h>

#define DIM      32000
#define NV4      (DIM / 4)
#define BLOCK    256
#define NWAVES   (BLOCK / 32)
#define NITER    50
#define ACT_CAP  128          // expected active-set size ~6 for N(0,1); fallback if exceeded

typedef __attribute__((ext_vector_type(4))) unsigned int v4u;
typedef __attribute__((ext_vector_type(8))) unsigned int v8u;

// ---------------- wave32 / block reductions ----------------
__device__ __forceinline__ float blockSum(float v, float* scratch) {
  const int lane = threadIdx.x & 31;
  const int w    = threadIdx.x >> 5;
#pragma unroll
  for (int o = 16; o > 0; o >>= 1) v += __shfl_xor(v, o, 32);
  __syncthreads();                 // protect scratch from any previous use
  if (lane == 0) scratch[w] = v;
  __syncthreads();
  if (w == 0) {
    float t = (lane < NWAVES) ? scratch[lane] : 0.0f;
#pragma unroll
    for (int o = 16; o > 0; o >>= 1) t += __shfl_xor(t, o, 32);
    if (lane == 0) scratch[0] = t;
  }
  __syncthreads();
  return scratch[0];
}

__device__ __forceinline__ float blockMax(float v, float* scratch) {
  const int lane = threadIdx.x & 31;
  const int w    = threadIdx.x >> 5;
#pragma unroll
  for (int o = 16; o > 0; o >>= 1) v = fmaxf(v, __shfl_xor(v, o, 32));
  __syncthreads();
  if (lane == 0) scratch[w] = v;
  __syncthreads();
  if (w == 0) {
    float t = (lane < NWAVES) ? scratch[lane] : -__builtin_inff();
#pragma unroll
    for (int o = 16; o > 0; o >>= 1) t = fmaxf(t, __shfl_xor(t, o, 32));
    if (lane == 0) scratch[0] = t;
  }
  __syncthreads();
  return scratch[0];
}

// partial Sum(relu(x - tau)) over this thread's strided slice of the LDS row
__device__ __forceinline__ float reluSumPartial(const float4* srv, float tau) {
  float s = 0.0f;
  for (int i = threadIdx.x; i < NV4; i += BLOCK) {
    float4 v = srv[i];
    s += fmaxf(v.x - tau, 0.0f) + fmaxf(v.y - tau, 0.0f) +
         fmaxf(v.z - tau, 0.0f) + fmaxf(v.w - tau, 0.0f);
  }
  return s;
}

__global__ void logsparsemax_bisect_kernel(const float* __restrict__ X,
                                           float* __restrict__ Y) {
  extern __shared__ __align__(16) float smem[];
  float* sr   = smem;                    // DIM floats: the row
  float* act  = smem + DIM;              // ACT_CAP floats: active set
  float* red  = act + ACT_CAP;           // 16 floats: reduction scratch
  int*   tcnt = (int*)(red + 16);        // BLOCK+1 ints: compaction prefix-scan

  const int    row = blockIdx.x;
  const float* Xr  = X + (size_t)row * DIM;
  float*       Yr  = Y + (size_t)row * DIM;

  // ---- Phase 1: TDM DMA the whole 128KB row into LDS (one instruction, wave 0)
  const int wid = __builtin_amdgcn_readfirstlane((int)threadIdx.x) >> 5;  // uniform -> scalar branch
  if (wid == 0) {
    unsigned long long ga = (unsigned long long)(uintptr_t)Xr & 0x01FFFFFFFFFFFFFFull;
    unsigned lds = (unsigned)(uintptr_t)sr;   // low 32 bits = LDS byte offset
    v4u g0;
    g0.x = 1u;                                   // count=1 (valid user descriptor)
    g0.y = lds;                                  // lds_addr
    g0.z = (unsigned)(ga & 0xFFFFFFFFu);         // global_addr[31:0]
    g0.w = (unsigned)(ga >> 32) | 0x80000000u;   // global_addr[56:32] | type=2 (bits 127:126)
    v8u g1;
    g1.s0 = 2u << 16;                            // data_size=4B; no multicast/pad/iterate
    g1.s1 = (unsigned)(DIM & 0xFFFF) << 16;      // tensor_dim0[15:0]  (bits 63:48)
    g1.s2 = (unsigned)(DIM >> 16) | (1u << 16);  // tensor_dim0[31:16] | tensor_dim1=1
    g1.s3 = (unsigned)DIM << 16;                 // tile_dim0 = 32000  (bits 127:112)
    g1.s4 = 1u;                                  // tile_dim1=1, tile_dim2=0
    g1.s5 = (unsigned)DIM;                       // tensor_dim0_stride[31:0]
    g1.s6 = (unsigned)(DIM & 0xFFFF) << 16;      // stride[47:32]=0 | tensor_dim1_stride[15:0]
    g1.s7 = 0u;
    asm volatile("tensor_load_to_lds %0, %1" :: "s"(g0), "s"(g1) : "memory");
    __builtin_amdgcn_s_wait_tensorcnt(0);
  }
  __syncthreads();   // DMA complete + visible to all waves

  const float4* srv = (const float4*)sr;

  // ---- Phase 2: row max
  float m = -__builtin_inff();
  for (int i = threadIdx.x; i < NV4; i += BLOCK) {
    float4 v = srv[i];
    m = fmaxf(m, fmaxf(fmaxf(v.x, v.y), fmaxf(v.z, v.w)));
  }
  const float mx     = blockMax(m, red);
  const float tauLo0 = mx - 1.0f;
  const float tauHi0 = mx - 1.0f / (float)DIM;

  // ---- Phase 3: classify. x > tauHi0 -> linear contributor (closed form);
  //      tauLo0 < x <= tauHi0 -> active set; x <= tauLo0 -> always zero.
  float sHi = 0.0f, cHi = 0.0f;
  int   myAct = 0;
  for (int i = threadIdx.x; i < NV4; i += BLOCK) {
    float4 v = srv[i];
    float xs[4] = {v.x, v.y, v.z, v.w};
#pragma unroll
    for (int k = 0; k < 4; ++k) {
      float x = xs[k];
      if (x > tauHi0)      { sHi += x; cHi += 1.0f; }
      else if (x > tauLo0) { ++myAct; }
    }
  }
  tcnt[threadIdx.x] = myAct;
  sHi = blockSum(sHi, red);
  cHi = blockSum(cHi, red);
  if (threadIdx.x == 0) {            // deterministic exclusive scan (once per row)
    int acc = 0;
#pragma unroll 1
    for (int t = 0; t < BLOCK; ++t) { int c = tcnt[t]; tcnt[t] = acc; acc += c; }
    tcnt[BLOCK] = acc;
  }
  __syncthreads();
  const int nAct = tcnt[BLOCK];

  if (nAct <= ACT_CAP) {             // deterministic compaction pass
    int k = tcnt[threadIdx.x];
    for (int i = threadIdx.x; i < NV4; i += BLOCK) {
      float4 v = srv[i];
      float xs[4] = {v.x, v.y, v.z, v.w};
#pragma unroll
      for (int c = 0; c < 4; ++c) {
        float x = xs[c];
        if (x <= tauHi0 && x > tauLo0) act[k++] = x;
      }
    }
  }
  __syncthreads();

  // ---- Phase 4: bisection (faithful to SparsemaxBisectFunction, alpha=2)
  float tauM, fM;
  if (nAct <= ACT_CAP) {
    // tiny active set: every thread redundantly runs the identical serial loop
    float fLo = sHi - cHi * tauLo0 - 1.0f;
    for (int j = 0; j < nAct; ++j) fLo += (act[j] - tauLo0);
    float tauLo = tauLo0, dm = tauHi0 - tauLo0;
    tauM = tauLo; fM = fLo;
    for (int it = 0; it < NITER; ++it) {
      dm  *= 0.5f;
      tauM = tauLo + dm;
      float s = sHi - cHi * tauM;
      for (int j = 0; j < nAct; ++j) s += fmaxf(act[j] - tauM, 0.0f);
      fM = s - 1.0f;
      if (fM * fLo >= 0.0f) tauLo = tauM;
    }
  } else {
    // robust fallback: full-row scans from LDS each iteration
    float fLo = blockSum(reluSumPartial(srv, tauLo0), red) - 1.0f;
    float tauLo = tauLo0, dm = tauHi0 - tauLo0;
    tauM = tauLo; fM = fLo;
    for (int it = 0; it < NITER; ++it) {
      dm  *= 0.5f;
      tauM = tauLo + dm;
      fM = blockSum(reluSumPartial(srv, tauM), red) - 1.0f;
      if (fM * fLo >= 0.0f) tauLo = tauM;
    }
  }
  const float sP = fM + 1.0f;        // sum(relu(x - tauM)) at final tauM (> 0)

  // ---- Phase 5: second normalization (elementwise divide-then-sum, as reference)
  float part = 0.0f;
  for (int i = threadIdx.x; i < NV4; i += BLOCK) {
    float4 v = srv[i];
    part += fmaxf(v.x - tauM, 0.0f) / sP + fmaxf(v.y - tauM, 0.0f) / sP +
            fmaxf(v.z - tauM, 0.0f) / sP + fmaxf(v.w - tauM, 0.0f) / sP;
  }
  const float s2 = blockSum(part, red);
  const float r  = 1.0f / (sP * s2);

  // ---- Phase 6: out = log(p / sP / s2); zero support -> -inf
  float4* Yv = (float4*)Yr;
  for (int i = threadIdx.x; i < NV4; i += BLOCK) {
    float4 v = srv[i];
    float4 o;
    o.x = logf(fmaxf(v.x - tauM, 0.0f) * r);
    o.y = logf(fmaxf(v.y - tauM, 0.0f) * r);
    o.z = logf(fmaxf(v.z - tauM, 0.0f) * r);
    o.w = logf(fmaxf(v.w - tauM, 0.0f) * r);
    Yv[i] = o;
  }
}

extern "C" void kernel_launch(void* const* d_in, const int* in_sizes, int n_in,
                              void* d_out, int out_size, void* d_ws, size_t ws_size,
                              hipStream_t stream) {
  (void)n_in; (void)out_size; (void)d_ws; (void)ws_size;
  const float* X = (const float*)d_in[0];
  float*       Y = (float*)d_out;
  const int nrows = in_sizes[0] / DIM;
  if (nrows <= 0) return;

  const size_t ldsBytes = (size_t)(DIM + ACT_CAP + 16) * sizeof(float) +
                          (size_t)(BLOCK + 1) * sizeof(int);
  // Allow >64KB dynamic LDS (ignore failure; AMD typically permits up to device max)
  (void)hipFuncSetAttribute((const void*)logsparsemax_bisect_kernel,
                            hipFuncAttributeMaxDynamicSharedMemorySize,
                            (int)ldsBytes);
  logsparsemax_bisect_kernel<<<dim3(nrows), dim3(BLOCK), ldsBytes, stream>>>(X, Y);
}